// BertNerHUB_10359461118299
// MI455X (gfx1250) — compile-verified
//
#include <hip/hip_runtime.h>
#include <hip/hip_bf16.h>
#include <stddef.h>

typedef __attribute__((ext_vector_type(2))) float v2f;
typedef __attribute__((ext_vector_type(8))) float v8f;

#define BB 64
#define SS 512
#define HH 1024
#define LL 9
#define WAVES_PER_BLOCK 8

// ---------------------------------------------------------------------------
// Kernel A: per-batch stable compaction indices + n_valid + zero row.
// One block per batch, 512 threads (one per token). Hillis-Steele scan in LDS.
// ---------------------------------------------------------------------------
__global__ __launch_bounds__(SS) void ner_compact(const int* __restrict__ mask,
                                                  int* __restrict__ idxbuf,
                                                  int* __restrict__ nvalid,
                                                  float* __restrict__ zrow) {
  __shared__ int sc[SS];
  const int b   = blockIdx.x;
  const int tid = threadIdx.x;
  const int m   = mask[b * SS + tid];
  sc[tid] = m;
  __syncthreads();
  for (int off = 1; off < SS; off <<= 1) {
    int v = 0;
    if (tid >= off) v = sc[tid - off];
    __syncthreads();
    if (tid >= off) sc[tid] += v;
    __syncthreads();
  }
  const int incl = sc[tid];                    // inclusive prefix count
  if (m) idxbuf[b * SS + (incl - 1)] = tid;    // stable: rank among valid
  if (tid == SS - 1) nvalid[b] = incl;
  if (b == 0) {                                // zero row for padded-tail lanes
    zrow[tid]      = 0.0f;
    zrow[tid + SS] = 0.0f;
  }
}

// ---------------------------------------------------------------------------
// Kernel B: WMMA f32 GEMM (gathered A rows x padded W) + bias + softmax.
// 256 threads = 8 waves; each wave owns one 16-token tile. 2048 tiles total.
// A: 16x4 f32 per k-step (lane M = lane&15; lanes<16 hold K0/K1, >=16 K2/K3).
// B: padded W (1024x16) in LDS; per k-step lane N = lane&15 reads rows
//    (4k + 2*khalf) and (4k + 2*khalf + 1)  -> conflict-free across banks.
// ---------------------------------------------------------------------------
__global__ __launch_bounds__(256) void ner_gemm_softmax(
    const float* __restrict__ seq, const float* __restrict__ W,
    const float* __restrict__ bias, const int* __restrict__ idxbuf,
    const int* __restrict__ nvalid, const float* __restrict__ zrow,
    float* __restrict__ out) {
  __shared__ float wlds[HH * 16];  // 64 KB, zero-padded columns 9..15

  const int tid = threadIdx.x;
  for (int i = tid; i < HH * 16; i += 256) {
    const int row = i >> 4;
    const int col = i & 15;
    wlds[i] = (col < LL) ? W[row * LL + col] : 0.0f;
  }
  __syncthreads();

  const int wave  = tid >> 5;
  const int lane  = tid & 31;
  const int n     = lane & 15;   // A row (M) for loads / B,C column (N)
  const int khalf = lane >> 4;   // which K half this lane holds

  const int tile = blockIdx.x * WAVES_PER_BLOCK + wave;  // 0..2047
  const int b    = tile >> 5;    // tile / (S/16)
  const int t    = tile & 31;

  const int  nv    = nvalid[b];
  const int  j     = t * 16 + n;            // compacted row owned as A's M=n
  const bool valid = (j < nv);
  const int  src   = valid ? idxbuf[b * SS + j] : 0;
  const float* arow =
      valid ? (seq + (((size_t)(b * SS + src)) << 10) + (khalf << 1))
            : (zrow + (khalf << 1));

  v8f acc = {};
#pragma unroll 4
  for (int kk = 0; kk < HH / 4; ++kk) {
    const float2 av = *(const float2*)(arow + (kk << 2));
    const int    wb = (((kk << 2) + (khalf << 1)) << 4) + n;
    v2f a;
    a.x = av.x;
    a.y = av.y;
    v2f w;
    w.x = wlds[wb];
    w.y = wlds[wb + 16];
    acc = __builtin_amdgcn_wmma_f32_16x16x4_f32(
        /*neg_a=*/false, a, /*neg_b=*/false, w,
        /*c_mod=*/(short)0, acc, /*reuse_a=*/false, /*reuse_b=*/false);
  }

  // Epilogue: bias + softmax over columns 0..8 within each 16-lane group.
  const float bn      = (n < LL) ? bias[n] : 0.0f;
  const int   baserow = b * SS + t * 16;
#pragma unroll
  for (int r = 0; r < 8; ++r) {
    const int   M = r + (khalf << 3);  // C/D: lanes>=16 hold M = r + 8
    float v = (n < LL) ? (acc[r] + bn) : -3.402823466e38f;
    float mx = v;
    mx = fmaxf(mx, __shfl_xor(mx, 1, 16));
    mx = fmaxf(mx, __shfl_xor(mx, 2, 16));
    mx = fmaxf(mx, __shfl_xor(mx, 4, 16));
    mx = fmaxf(mx, __shfl_xor(mx, 8, 16));
    const float e = __expf(v - mx);  // masked lanes -> exp(-huge) = 0
    float s = e;
    s += __shfl_xor(s, 1, 16);
    s += __shfl_xor(s, 2, 16);
    s += __shfl_xor(s, 4, 16);
    s += __shfl_xor(s, 8, 16);
    if (n < LL) out[(size_t)(baserow + M) * LL + n] = e / s;
  }
}

// ---------------------------------------------------------------------------
// Launch. Inputs: [0] sequence_output f32 [64,512,1024], [1] valid_mask i32
// [64,512], [2] W f32 [1024,9], [3] b f32 [9]. Output f32 [64,512,9].
// ---------------------------------------------------------------------------
extern "C" void kernel_launch(void* const* d_in, const int* in_sizes, int n_in,
                              void* d_out, int out_size, void* d_ws,
                              size_t ws_size, hipStream_t stream) {
  const float* seq   = (const float*)d_in[0];
  const int*   vmask = (const int*)d_in[1];
  const float* W     = (const float*)d_in[2];
  const float* bias  = (const float*)d_in[3];
  float*       out   = (float*)d_out;

  char* ws     = (char*)d_ws;
  int*  idxbuf = (int*)ws;                                 // 64*512*4 B
  int*  nvld   = (int*)(ws + (size_t)BB * SS * 4);         // 256 B
  float* zrow  = (float*)(ws + (size_t)BB * SS * 4 + 1024);// 4 KB zero row

  ner_compact<<<BB, SS, 0, stream>>>(vmask, idxbuf, nvld, zrow);

  const int tiles  = BB * (SS / 16);                 // 2048
  const int blocks = tiles / WAVES_PER_BLOCK;        // 256
  ner_gemm_softmax<<<blocks, 256, 0, stream>>>(seq, W, bias, idxbuf, nvld,
                                               zrow, out);
}